// RandomDeformSketch_33062658245456
// MI455X (gfx1250) — compile-verified
//
#include <hip/hip_runtime.h>
#include <hip/hip_bf16.h>

// Constants from the reference
#define S_DIM 1024
#define G_DIM 10
#define F_RAD 5
#define B_DIM 8
#define C_DIM 3
#define KW 11              // 2*F+1
#define NT 112             // padded N for T (7 tiles of 16)

typedef __attribute__((ext_vector_type(2))) float v2f;
typedef __attribute__((ext_vector_type(8))) float v8f;

// ---------------------------------------------------------------------------
// Kernel 1: build the 100x112 linear operator T such that
//   smooth(o).flatten() = X(16x100) * T   (edge-padded 11x11 conv collapsed
//   into per-source-cell accumulated weights). Each output column c is owned
//   by exactly one thread -> no atomics needed.
// ---------------------------------------------------------------------------
__global__ void build_T_kernel(const float* __restrict__ w, float* __restrict__ T) {
    int tid = threadIdx.x;
    for (int i = tid; i < G_DIM * G_DIM * NT; i += blockDim.x) T[i] = 0.0f;
    __syncthreads();
    for (int c = tid; c < G_DIM * G_DIM; c += blockDim.x) {
        int i = c / G_DIM, j = c % G_DIM;
        for (int k = 0; k < KW * KW; ++k) {
            int a = k / KW, b = k % KW;
            int ii = min(max(i + a - F_RAD, 0), G_DIM - 1);
            int jj = min(max(j + b - F_RAD, 0), G_DIM - 1);
            T[(ii * G_DIM + jj) * NT + c] += w[k];
        }
    }
}

// ---------------------------------------------------------------------------
// Kernel 2: O = X * T via V_WMMA_F32_16X16X4_F32 (fp32 matrix pipe).
//   M = 16 rows = (batch, {x,y}) offset fields, r = b*2 + ch
//   K = 100 source cells (25 chunks of 4), N = 100 outputs (7 tiles of 16)
// One wave per N-tile -> 1 block of 224 threads (7 full wave32s, EXEC all-1s).
// Result scaled by max_offset and clipped, written to gsm[16][100].
// ---------------------------------------------------------------------------
__global__ void smooth_wmma_kernel(const float* __restrict__ ox,
                                   const float* __restrict__ oy,
                                   const float* __restrict__ T,
                                   const int*   __restrict__ mm,
                                   float* __restrict__ gsm) {
    const int lane = threadIdx.x & 31;
    const int tile = threadIdx.x >> 5;        // 0..6  -> N tile
    const int row  = lane & 15;               // M row for A, N col for B/D
    const int half = lane >> 4;               // lane half

    // A row r = b*2 + ch : ch 0 -> offset_x, ch 1 -> offset_y
    const float* xrow = ((row & 1) ? oy : ox) + (row >> 1) * (G_DIM * G_DIM);
    const int ncol = tile * 16 + row;         // global N column (0..111)

    v8f acc = {};
    for (int kk = 0; kk < 100; kk += 4) {
        // A 16x4 f32: lanes 0-15 hold K=kk+0,kk+1 ; lanes 16-31 hold K=kk+2,kk+3
        v2f a;
        a.x = xrow[kk + 2 * half + 0];
        a.y = xrow[kk + 2 * half + 1];
        // B 4x16 f32: VGPR i, lane-half h holds K = kk + 2*h + i, col = lane&15
        v2f b;
        b.x = T[(kk + 2 * half + 0) * NT + ncol];
        b.y = T[(kk + 2 * half + 1) * NT + ncol];
        acc = __builtin_amdgcn_wmma_f32_16x16x4_f32(
            /*neg_a=*/false, a, /*neg_b=*/false, b,
            /*c_mod=*/(short)0, acc, /*reuse_a=*/false, /*reuse_b=*/false);
    }

    const float max_offset = 2.0f * (float)mm[0] / (float)S_DIM;
    if (ncol < G_DIM * G_DIM) {
        #pragma unroll
        for (int v = 0; v < 8; ++v) {
            int m = v + 8 * half;             // D: M = v + 8*half, N = lane&15
            float val = acc[v] * max_offset;
            val = fminf(fmaxf(val, -max_offset), max_offset);
            gsm[m * (G_DIM * G_DIM) + ncol] = val;
        }
    }
}

// ---------------------------------------------------------------------------
// Kernel 3: upsample offsets + grid_sample. One thread = one (b,y,x) pixel,
// all 3 channels. Per-batch 2x10x10 smoothed-offset table cached in LDS.
// Memory-bound gather: ~12 cached loads + 3 streaming stores per thread.
// ---------------------------------------------------------------------------
__global__ void deform_sample_kernel(const float* __restrict__ xin,
                                     const float* __restrict__ gsm,
                                     float* __restrict__ out) {
    __shared__ float sg[2 * G_DIM * G_DIM];   // [ch][10][10] for this batch

    const int x = blockIdx.x * blockDim.x + threadIdx.x;
    const int y = blockIdx.y;
    const int b = blockIdx.z;

    if (threadIdx.x < 2 * G_DIM * G_DIM)
        sg[threadIdx.x] = gsm[b * 2 * G_DIM * G_DIM + threadIdx.x];
    __syncthreads();

    // --- bilinear upsample of offset grid (10 -> 1024), per reference ---
    const float scale = (float)G_DIM / (float)S_DIM;
    float sx = fmaxf(((float)x + 0.5f) * scale - 0.5f, 0.0f);
    float sy = fmaxf(((float)y + 0.5f) * scale - 0.5f, 0.0f);
    int ix0 = min((int)sx, G_DIM - 1);        // sx >= 0 -> trunc == floor
    int iy0 = min((int)sy, G_DIM - 1);
    int ix1 = min(ix0 + 1, G_DIM - 1);
    int iy1 = min(iy0 + 1, G_DIM - 1);
    float wxg = sx - (float)ix0;
    float wyg = sy - (float)iy0;

    // rows first, then cols (matches reference fp order)
    float gxv, gyv;
    {
        const float* g0 = sg;                       // channel 0 (x offset)
        float t0 = g0[iy0 * G_DIM + ix0] * (1.0f - wyg) + g0[iy1 * G_DIM + ix0] * wyg;
        float t1 = g0[iy0 * G_DIM + ix1] * (1.0f - wyg) + g0[iy1 * G_DIM + ix1] * wyg;
        gxv = t0 * (1.0f - wxg) + t1 * wxg;
        const float* g1 = sg + G_DIM * G_DIM;       // channel 1 (y offset)
        float u0 = g1[iy0 * G_DIM + ix0] * (1.0f - wyg) + g1[iy1 * G_DIM + ix0] * wyg;
        float u1 = g1[iy0 * G_DIM + ix1] * (1.0f - wyg) + g1[iy1 * G_DIM + ix1] * wyg;
        gyv = u0 * (1.0f - wxg) + u1 * wxg;
    }

    // --- identity grid + clip ---
    const float inv = 1.0f / (float)(S_DIM - 1);
    float px = (float)x * inv * 2.0f - 1.0f;
    float py = (float)y * inv * 2.0f - 1.0f;
    float gx = fminf(fmaxf(gxv + px, -1.0f), 1.0f);
    float gy = fminf(fmaxf(gyv + py, -1.0f), 1.0f);

    // --- bilinear grid sample with zero padding ---
    float ixf = ((gx + 1.0f) * (float)S_DIM - 1.0f) * 0.5f;
    float iyf = ((gy + 1.0f) * (float)S_DIM - 1.0f) * 0.5f;
    int x0 = (int)floorf(ixf);
    int y0 = (int)floorf(iyf);
    int x1 = x0 + 1;
    int y1 = y0 + 1;
    float wx = ixf - (float)x0;
    float wy = iyf - (float)y0;

    int cx0 = min(max(x0, 0), S_DIM - 1);
    int cx1 = min(max(x1, 0), S_DIM - 1);
    int cy0 = min(max(y0, 0), S_DIM - 1);
    int cy1 = min(max(y1, 0), S_DIM - 1);
    float mx0 = (x0 >= 0 && x0 < S_DIM) ? 1.0f : 0.0f;
    float mx1 = (x1 >= 0 && x1 < S_DIM) ? 1.0f : 0.0f;
    float my0 = (y0 >= 0 && y0 < S_DIM) ? 1.0f : 0.0f;
    float my1 = (y1 >= 0 && y1 < S_DIM) ? 1.0f : 0.0f;

    float w00 = my0 * mx0 * (1.0f - wy) * (1.0f - wx);
    float w01 = my0 * mx1 * (1.0f - wy) * wx;
    float w10 = my1 * mx0 * wy * (1.0f - wx);
    float w11 = my1 * mx1 * wy * wx;

    const size_t r0 = (size_t)cy0 * S_DIM;
    const size_t r1 = (size_t)cy1 * S_DIM;

    #pragma unroll
    for (int c = 0; c < C_DIM; ++c) {
        const float* img = xin + ((size_t)b * C_DIM + c) * (size_t)S_DIM * S_DIM;
        float v = img[r0 + cx0] * w00 + img[r0 + cx1] * w01
                + img[r1 + cx0] * w10 + img[r1 + cx1] * w11;
        out[(((size_t)b * C_DIM + c) * S_DIM + y) * S_DIM + x] = v;
    }
}

// ---------------------------------------------------------------------------
extern "C" void kernel_launch(void* const* d_in, const int* in_sizes, int n_in,
                              void* d_out, int out_size, void* d_ws, size_t ws_size,
                              hipStream_t stream) {
    const float* x        = (const float*)d_in[0];   // (8,3,1024,1024)
    const float* offset_x = (const float*)d_in[1];   // (8,1,10,10)
    const float* offset_y = (const float*)d_in[2];   // (8,1,10,10)
    const float* weight   = (const float*)d_in[3];   // (1,1,11,11)
    const int*   max_move = (const int*)d_in[4];     // scalar 50

    float* T   = (float*)d_ws;                            // 100 x 112 floats
    float* gsm = T + G_DIM * G_DIM * NT;                  // 16 x 100 floats

    build_T_kernel<<<1, 128, 0, stream>>>(weight, T);

    smooth_wmma_kernel<<<1, 7 * 32, 0, stream>>>(offset_x, offset_y, T, max_move, gsm);

    dim3 block(256);
    dim3 grid(S_DIM / 256, S_DIM, B_DIM);
    deform_sample_kernel<<<grid, block, 0, stream>>>(x, gsm, (float*)d_out);
}